// GradientRefinementModule_44160853737572
// MI455X (gfx1250) — compile-verified
//
#include <hip/hip_runtime.h>
#include <hip/hip_bf16.h>

#define SIGLEN          2048
#define ROWS_PER_BLOCK  8          // 8 rows -> 64 KB LDS tile, 24/32 lanes refine
#define THREADS         32         // single wave32 workgroup: barriers are S_NOP
#define MAX_ITER        15

typedef unsigned int u32x4 __attribute__((ext_vector_type(4)));
typedef unsigned int u32x8 __attribute__((ext_vector_type(8)));

extern __shared__ float lds_sig[];   // ROWS_PER_BLOCK * SIGLEN floats (64 KB)

__device__ __forceinline__ float sample_row(const float* s, float pos) {
    float p  = pos * 2047.0f;
    p        = fminf(fmaxf(p, 0.0f), 2047.0f);
    float fl = floorf(p);
    int   il = (int)fl;
    int   ir = (int)ceilf(p);
    float wr = p - fl;
    float vl = s[il];                 // ds_load_b32 from LDS
    float vr = s[ir];
    return (1.0f - wr) * vl + wr * vr;
}

__global__ void __launch_bounds__(THREADS)
grad_refine_tdm_kernel(const float* __restrict__ preds,
                       const float* __restrict__ signals,
                       float* __restrict__ out,
                       int batch) {
    const int row0 = blockIdx.x * ROWS_PER_BLOCK;
    int rows_here = batch - row0;
    if (rows_here <= 0) return;
    if (rows_here > ROWS_PER_BLOCK) rows_here = ROWS_PER_BLOCK;
    const unsigned n_elems = (unsigned)rows_here * SIGLEN;   // <= 16384, fits tile_dim0

    // ---- Kick off the per-element position load early (LOADcnt, independent of TENSORcnt)
    const int tid    = (int)threadIdx.x;
    const int r      = tid / 3;
    const int j      = tid - r * 3;
    const bool live  = (tid < rows_here * 3);
    const int  gidx  = (row0 + r) * 3 + j;
    float pos = live ? preds[gidx] : 0.0f;

    // ---------------- TDM: one tensor_load_to_lds streams the whole 64 KB tile ----------
    // Single wave per workgroup, so no gating needed (TDM ignores EXEC anyway).
    {
        unsigned long long gaddr =
            (unsigned long long)(const void*)(signals + (unsigned long long)row0 * SIGLEN);
        // Flat shared-space pointers carry the LDS byte offset in bits [31:0].
        unsigned lds_base = (unsigned)(unsigned long long)(void*)&lds_sig[0];

        // D# group 0 (128b): count=1 | lds_addr | global_addr[56:0] | type=2
        u32x4 g0 = {0u, 0u, 0u, 0u};
        g0[0] = 1u;                                            // count = 1 valid descriptor
        g0[1] = lds_base;                                      // lds_addr (bytes)
        g0[2] = (unsigned)gaddr;                               // global_addr[31:0]
        g0[3] = (unsigned)((gaddr >> 32) & 0x1FFFFFFull)       // global_addr[56:32]
              | (2u << 30);                                    // type = 2 ("image")

        // D# group 1 (256b): data_size=4B, 1-D tile covering the whole tensor
        u32x8 g1 = {0u,0u,0u,0u,0u,0u,0u,0u};
        g1[0] = (2u << 16);                                    // data_size = 2 -> 4 bytes
        g1[1] = (n_elems & 0xFFFFu) << 16;                     // tensor_dim0[15:0]  @ bits 63:48
        g1[2] = ((n_elems >> 16) & 0xFFFFu)                    // tensor_dim0[31:16] @ bits 79:64
              | (1u << 16);                                    // tensor_dim1 = 1
        g1[3] = (n_elems & 0xFFFFu) << 16;                     // tile_dim0 @ bits 127:112
        g1[4] = 0u;                                            // tile_dim1=0, tile_dim2=0 (1-D)
        g1[5] = n_elems;                                       // tensor_dim0_stride[31:0]
        g1[6] = 0u;
        g1[7] = 0u;

        asm volatile("tensor_load_to_lds %0, %1"
                     :: "s"(g0), "s"(g1)
                     : "memory");
        __builtin_amdgcn_s_wait_tensorcnt(0);                  // DMA complete -> LDS valid
    }
    __syncthreads();   // single-wave workgroup: compiles to S_NOP, kept for shape-safety

    // ---------------- Refinement: one lane per (row, position) element ------------------
    if (live) {
        const float* srow = &lds_sig[r * SIGLEN];

        float step_size = 0.01f;     // BASE_STEP_SIZE, decays 0.9x per iteration
        float max_step  = 0.2f;      // 0.2 * 0.9^it

        const float eps_s[3] = {0.001f, 0.003f, 0.01f};
        const float eps_w[3] = {0.5f, 0.3f, 0.2f};

        // The reference's stop condition is a batch-global max that never fires within
        // 15 iterations (step_size >= 0.00206 > MIN_STEP, and clip-saturated Newton
        // steps keep the global max position change above POS_TOL), so run all 15.
        #pragma unroll 1
        for (int it = 0; it < MAX_ITER; ++it) {
            float g = 0.0f, c = 0.0f;
            #pragma unroll
            for (int k = 0; k < 3; ++k) {
                const float eps = eps_s[k];
                const float w   = eps_w[k];
                float pc = fminf(fmaxf(pos, eps), 1.0f - eps);
                float v  = sample_row(srow, pc);
                float vl = sample_row(srow, pc - eps);
                float vr = sample_row(srow, pc + eps);
                g += w * (vr - vl) / (2.0f * eps);
                c += w * (vr + vl - 2.0f * v) / (eps * eps);
            }
            c = fminf(fmaxf(c, -1000.0f), 1000.0f);
            float step = -g / (fabsf(c) + 1e-6f);
            step = fminf(fmaxf(step, -max_step), max_step);
            pos  = fminf(fmaxf(pos + step_size * step, 0.0f), 1.0f);
            step_size *= 0.9f;
            max_step  *= 0.9f;
        }
        out[gidx] = pos;             // returned "best_positions" == final positions
    }
}

extern "C" void kernel_launch(void* const* d_in, const int* in_sizes, int n_in,
                              void* d_out, int out_size, void* d_ws, size_t ws_size,
                              hipStream_t stream) {
    const float* preds   = (const float*)d_in[0];   // (batch, 3) float32
    const float* signals = (const float*)d_in[1];   // (batch, 2048) float32
    float* out = (float*)d_out;                     // (batch, 3) float32

    const int batch = in_sizes[0] / 3;
    const int grid  = (batch + ROWS_PER_BLOCK - 1) / ROWS_PER_BLOCK;
    const size_t shmem = (size_t)ROWS_PER_BLOCK * SIGLEN * sizeof(float);  // 64 KB

    grad_refine_tdm_kernel<<<grid, THREADS, shmem, stream>>>(preds, signals, out, batch);
}